// Refinement_79431125172537
// MI455X (gfx1250) — compile-verified
//
#include <hip/hip_runtime.h>

typedef float v2f __attribute__((ext_vector_type(2)));
typedef float v8f __attribute__((ext_vector_type(8)));

#define N_PTS   16384
#define DIMS    64
#define MARGIN  1.0f
#define TILE    64                      // 64x64 output per workgroup
#define NB      (N_PTS / TILE)          // 256 tile-blocks per dim
#define NHPART  (NB * NB)               // 65536 hinge partials
#define NEPART  64                      // loss_e partials (16384/256 blocks)

// -------- Kernel 1: sq[i] = |t_i|^2  and  loss_e block partials --------
__global__ __launch_bounds__(256) void k_rows(const float* __restrict__ t,
                                              const float* __restrict__ cat,
                                              const int*   __restrict__ cid,
                                              float* __restrict__ sq,
                                              float* __restrict__ epart) {
    __shared__ float red[256];
    const int i = blockIdx.x * 256 + threadIdx.x;
    const float4* tr = (const float4*)(t + (size_t)i * DIMS);
    const float4* cr = (const float4*)(cat + (size_t)cid[i] * DIMS);
    float s = 0.f, e = 0.f;
#pragma unroll
    for (int q = 0; q < DIMS / 4; ++q) {
        float4 a = tr[q];
        float4 b = cr[q];
        s += a.x * a.x + a.y * a.y + a.z * a.z + a.w * a.w;
        float dx = a.x - b.x, dy = a.y - b.y, dz = a.z - b.z, dw = a.w - b.w;
        e += dx * dx + dy * dy + dz * dz + dw * dw;
    }
    sq[i] = s;
    red[threadIdx.x] = e;
    __syncthreads();
    for (int off = 128; off > 0; off >>= 1) {
        if (threadIdx.x < off) red[threadIdx.x] += red[threadIdx.x + off];
        __syncthreads();
    }
    if (threadIdx.x == 0) epart[blockIdx.x] = red[0];
}

// -------- Kernel 2: fused A*A^T (WMMA f32 16x16x4) + hinge epilogue --------
// Each 256-thread block (8 waves) computes a 64x64 tile of the distance
// matrix. Wave w owns 16x16 sub-tiles (tm0, tn) and (tm0+2, tn), which share
// the same B fragment.
__global__ __launch_bounds__(256) void k_pair(const float* __restrict__ t,
                                              const int*   __restrict__ cid,
                                              const float* __restrict__ sq,
                                              float* __restrict__ hpart) {
    __shared__ float red[256];
    const int lane  = threadIdx.x & 31;
    const int wave  = threadIdx.x >> 5;
    const int i0    = blockIdx.y * TILE;
    const int j0    = blockIdx.x * TILE;
    const int tm0   = wave >> 2;            // 0..1  (row sub-tile)
    const int tn    = wave & 3;             // 0..3  (col sub-tile)
    const int lrow  = lane & 15;
    const int halfk = (lane >> 4) * 2;      // K offset 0 or 2 per A/B layout

    // A fragments: row = i-block row, 2 consecutive K values per lane.
    const float* arow0 = t + (size_t)(i0 + tm0 * 16 + lrow) * DIMS + halfk;
    const float* arow1 = arow0 + (size_t)32 * DIMS;     // tm0+2 sub-tile
    // B = A^T of the j-block: same per-lane pattern with row -> column index.
    const float* brow  = t + (size_t)(j0 + tn * 16 + lrow) * DIMS + halfk;

    v8f acc0 = {};
    v8f acc1 = {};
#pragma unroll
    for (int kk = 0; kk < DIMS; kk += 4) {
        v2f a0 = *(const v2f*)(arow0 + kk);
        v2f a1 = *(const v2f*)(arow1 + kk);
        v2f b  = *(const v2f*)(brow  + kk);
        acc0 = __builtin_amdgcn_wmma_f32_16x16x4_f32(false, a0, false, b,
                                                     (short)0, acc0, false, false);
        acc1 = __builtin_amdgcn_wmma_f32_16x16x4_f32(false, a1, false, b,
                                                     (short)0, acc1, false, false);
    }

    // Epilogue. C layout: VGPR r holds (M = r + 8*(lane>>4), N = lane&15).
    const int j    = j0 + tn * 16 + lrow;
    const float sj = sq[j];
    const int   cj = cid[j];
    const int ibase0 = i0 + tm0 * 16 + ((lane >> 4) ? 8 : 0);
    const int ibase1 = ibase0 + 32;

    float hsum = 0.f;
#pragma unroll
    for (int r = 0; r < 8; ++r) {
        {
            const int i = ibase0 + r;
            const float d = sq[i] + sj - 2.0f * acc0[r];
            hsum += (cid[i] == cj) ? MARGIN : fmaxf(MARGIN - d, 0.0f);
        }
        {
            const int i = ibase1 + r;
            const float d = sq[i] + sj - 2.0f * acc1[r];
            hsum += (cid[i] == cj) ? MARGIN : fmaxf(MARGIN - d, 0.0f);
        }
    }

    red[threadIdx.x] = hsum;
    __syncthreads();
    for (int off = 128; off > 0; off >>= 1) {
        if (threadIdx.x < off) red[threadIdx.x] += red[threadIdx.x + off];
        __syncthreads();
    }
    if (threadIdx.x == 0) hpart[blockIdx.y * NB + blockIdx.x] = red[0];
}

// -------- Kernel 3: deterministic final reduction --------
__global__ __launch_bounds__(256) void k_final(const float* __restrict__ epart,
                                               const float* __restrict__ hpart,
                                               float* __restrict__ out) {
    __shared__ float rh[256];
    __shared__ float re[256];
    float h = 0.f;
    for (int idx = threadIdx.x; idx < NHPART; idx += 256) h += hpart[idx];
    float e = (threadIdx.x < NEPART) ? epart[threadIdx.x] : 0.f;
    rh[threadIdx.x] = h;
    re[threadIdx.x] = e;
    __syncthreads();
    for (int off = 128; off > 0; off >>= 1) {
        if (threadIdx.x < off) {
            rh[threadIdx.x] += rh[threadIdx.x + off];
            re[threadIdx.x] += re[threadIdx.x + off];
        }
        __syncthreads();
    }
    if (threadIdx.x == 0) out[0] = re[0] + rh[0] / (float)N_PTS;
}

extern "C" void kernel_launch(void* const* d_in, const int* in_sizes, int n_in,
                              void* d_out, int out_size, void* d_ws, size_t ws_size,
                              hipStream_t stream) {
    const float* t   = (const float*)d_in[0];   // target_reps   [N, D] f32
    const float* cat = (const float*)d_in[1];   // category_reps [K, D] f32
    const int*   cid = (const int*)d_in[2];     // cluster_ids   [N]
    float* out = (float*)d_out;

    float* ws    = (float*)d_ws;
    float* sq    = ws;                  // N_PTS floats
    float* epart = ws + N_PTS;          // NEPART floats
    float* hpart = epart + NEPART;      // NHPART floats

    k_rows<<<N_PTS / 256, 256, 0, stream>>>(t, cat, cid, sq, epart);
    k_pair<<<dim3(NB, NB), 256, 0, stream>>>(t, cid, sq, hpart);
    k_final<<<1, 256, 0, stream>>>(epart, hpart, out);
}